// Predictor_12893491822699
// MI455X (gfx1250) — compile-verified
//
#include <hip/hip_runtime.h>

// Edge-wise dot product via V_WMMA_F32_16X16X4_F32 on gfx1250 (wave32).
// 16 edges per wave. Each lane loads b128 (float4) per matrix per 8-column
// step; a K-permutation (legal for dot products) lets one b128 feed two
// WMMA ops, halving vmem instruction count vs. the b64 version.
// Full fp32 precision (matches fp32 reference).

typedef float v2f __attribute__((ext_vector_type(2)));
typedef float v4f __attribute__((ext_vector_type(4)));
typedef float v8f __attribute__((ext_vector_type(8)));

#define DIM 128
#define WAVES_PER_BLOCK 8

__global__ __launch_bounds__(256) void edge_dot_wmma_f32(
    const float* __restrict__ user_h,
    const float* __restrict__ game_h,
    const int*   __restrict__ src_idx,
    const int*   __restrict__ dst_idx,
    float*       __restrict__ out,
    int          num_edges)
{
    const int lane  = threadIdx.x & 31;
    const int wave  = threadIdx.x >> 5;
    const int tile  = blockIdx.x * WAVES_PER_BLOCK + wave;  // 16 edges / tile
    const int e0    = tile * 16;

    // Fragment row (edge within tile) and lane-half (K selector).
    const int m     = lane & 15;
    const int khalf = lane >> 4;      // 0: lanes 0-15 (K0,K1), 1: lanes 16-31 (K2,K3)

    int e  = e0 + m;
    int ec = (e < num_edges) ? e : (num_edges - 1);   // clamp, keep EXEC full

    const int ru = src_idx[ec];
    const int rv = dst_idx[ec];

    // Each lane-half covers a disjoint 4-column span per 8-column step.
    const float* up = user_h + (size_t)ru * DIM + 4 * khalf;
    const float* vp = game_h + (size_t)rv * DIM + 4 * khalf;

    v8f c = {};
    #pragma unroll
    for (int kb = 0; kb < DIM; kb += 8) {
        v4f a4 = *(const v4f*)(up + kb);   // cols kb+4*khalf .. +3 of U row
        v4f b4 = *(const v4f*)(vp + kb);   // cols kb+4*khalf .. +3 of V row

        // WMMA #1 consumes logical columns {kb, kb+1, kb+4, kb+5}
        v2f a01 = __builtin_shufflevector(a4, a4, 0, 1);
        v2f b01 = __builtin_shufflevector(b4, b4, 0, 1);
        c = __builtin_amdgcn_wmma_f32_16x16x4_f32(
                false, a01, false, b01, (short)0, c, false, false);

        // WMMA #2 consumes logical columns {kb+2, kb+3, kb+6, kb+7}
        v2f a23 = __builtin_shufflevector(a4, a4, 2, 3);
        v2f b23 = __builtin_shufflevector(b4, b4, 2, 3);
        c = __builtin_amdgcn_wmma_f32_16x16x4_f32(
                false, a23, false, b23, (short)0, c, false, false);
    }

    // Extract diagonal D[m][m]:
    //   m in 0..7  -> VGPR m,   lane m
    //   m in 8..15 -> VGPR m-8, lane m+16
    int outm = -1, cidx = 0;
    if (lane < 8)        { outm = lane;     cidx = lane;      }
    else if (lane >= 24) { outm = lane - 8; cidx = lane - 24; }

    float val = 0.0f;
    #pragma unroll
    for (int i = 0; i < 8; ++i)
        val = (cidx == i) ? c[i] : val;

    int eo = e0 + outm;
    if (outm >= 0 && eo < num_edges)
        out[eo] = val;
}

extern "C" void kernel_launch(void* const* d_in, const int* in_sizes, int n_in,
                              void* d_out, int out_size, void* d_ws, size_t ws_size,
                              hipStream_t stream) {
    const float* user_h  = (const float*)d_in[0];
    const float* game_h  = (const float*)d_in[1];
    const int*   src_idx = (const int*)d_in[2];
    const int*   dst_idx = (const int*)d_in[3];
    float*       out     = (float*)d_out;

    const int num_edges = in_sizes[2];                 // E
    const int tiles     = (num_edges + 15) / 16;       // one wave per 16 edges
    const int blocks    = (tiles + WAVES_PER_BLOCK - 1) / WAVES_PER_BLOCK;

    edge_dot_wmma_f32<<<blocks, 32 * WAVES_PER_BLOCK, 0, stream>>>(
        user_h, game_h, src_idx, dst_idx, out, num_edges);
}